// ADAM_SINDy_MODEL_64476049048182
// MI455X (gfx1250) — compile-verified
//
#include <hip/hip_runtime.h>
#include <stdint.h>

// Problem constants (from the reference)
#define N_PROLIF 12
#define N_LIN    18
#define THREADS  256
#define PPT      2                    // points per thread
#define PPB      (THREADS * PPT)      // 512 points per block
#define CH       22                   // f32 channels per point
#define REC_B    (CH * 4)             // 88 B record; stride 22 dw is conflict-free
                                      // for wave32: bank = 2*(11*t mod 32), bijection
#define HALF_B   (THREADS * REC_B)    // 22528 B: second point = same vaddr + imm offset
#define BLK_B    (PPB * REC_B)        // 45056 B staged per block = 11 x 4096

#if __has_builtin(__builtin_amdgcn_tanhf)
#define USE_TANH 1
#else
#define USE_TANH 0
#endif

// Resolved, pre-gated parameter block (written once per launch by setup_params).
struct __align__(16) Params {
    float k1, nthinv, k2, coef0;    // nthinv = -1/theta
    int   p_xoff0[N_PROLIF];        // byte offset of x-operand 0 in point record
    int   p_xoff1[N_PROLIF];
    int   p_poff [N_PROLIF];        // byte offset of protein operand
    float p_coef [N_PROLIF];        // gated a[1+t]
    int   l_xoff [N_LIN];
    int   l_poff [N_LIN];
    float l_ss   [N_LIN];           // steady_state[lin_hill_idx[t,1]] resolved
    float l_sc   [N_LIN];           // tanh: 250*sign ; exp: 500*sign
    float l_coef [N_LIN];           // tanh: 0.5*gated a[13+t] ; exp: gated a[13+t]
};

// Record layout after compute-phase rewrite:
//   my[0] = xsum (con consumed first), my[1..3] = x0..x2, my[4..21] = proteins
__device__ __forceinline__ int xoff_of(int i) { return (i < 3) ? (4 + 4 * i) : 0; }

__global__ void setup_params(const float* __restrict__ a,
                             const float* __restrict__ ss,
                             const float* __restrict__ sgn,
                             const float* __restrict__ K1,
                             const float* __restrict__ theta,
                             const float* __restrict__ K2,
                             const int*   __restrict__ pidx,   // (12,3)
                             const int*   __restrict__ lidx,   // (18,2)
                             const unsigned char* __restrict__ sp, // (31,) bool
                             Params* __restrict__ P)
{
    if (threadIdx.x != 0 || blockIdx.x != 0) return;
    P->k1     = K1[0];
    P->k2     = K2[0];
    P->nthinv = -1.0f / theta[0];

    auto gated = [&](int k) -> float {
        float ak = a[k];
        bool keep = sp[k] ? (ak >= 0.0f) : (ak <= 0.0f);
        return keep ? ak : 0.0f;
    };
    P->coef0 = gated(0);

    for (int t = 0; t < N_PROLIF; ++t) {
        P->p_xoff0[t] = xoff_of(pidx[3 * t + 0]);
        P->p_xoff1[t] = xoff_of(pidx[3 * t + 1]);
        P->p_poff [t] = 16 + 4 * pidx[3 * t + 2];   // proteins = channels 4..21
        P->p_coef [t] = gated(1 + t);
    }
    for (int t = 0; t < N_LIN; ++t) {
        const int ip = lidx[2 * t + 1];
        P->l_xoff[t] = xoff_of(lidx[2 * t + 0]);
        P->l_poff[t] = 16 + 4 * ip;
        P->l_ss  [t] = ss[ip];
#if USE_TANH
        P->l_sc  [t] = 250.0f * sgn[t];             // z/2 for tanh form
        P->l_coef[t] = 0.5f * gated(13 + t);        // fold sigmoid 0.5 into coef
#else
        P->l_sc  [t] = 500.0f * sgn[t];
        P->l_coef[t] = gated(13 + t);
#endif
    }
}

// CDNA5 async global->LDS: IOFFSET applies to BOTH the LDS destination and the
// global source, so one base-VGPR pair + immediates covers the whole block.
#define ASYNC_B128(LADDR, GOFF, SBASE, OFF)                                    \
    asm volatile("global_load_async_to_lds_b128 %0, %1, %2 offset:" #OFF      \
                 :: "v"(LADDR), "v"(GOFF), "s"(SBASE) : "memory")

__global__ __launch_bounds__(THREADS) void sindy_main(
    const float*  __restrict__ cand,   // (B,T,22)
    const Params* __restrict__ P,
    float*        __restrict__ out)    // (B,T)
{
    __shared__ __align__(16) float smem[PPB * CH];   // 45056 B
    const int tid = threadIdx.x;

    // ---------------- Stage: 45056 B global -> LDS, fully coalesced ----------
    const float*   gsrc = cand + (size_t)blockIdx.x * (PPB * CH);
    const uint32_t lds0 = (uint32_t)(uintptr_t)(&smem[0]);
    const uint32_t g16  = (uint32_t)tid * 16u;       // 16 B chunk per lane
    const uint32_t l16  = lds0 + g16;
    ASYNC_B128(l16, g16, gsrc, 0);
    ASYNC_B128(l16, g16, gsrc, 4096);
    ASYNC_B128(l16, g16, gsrc, 8192);
    ASYNC_B128(l16, g16, gsrc, 12288);
    ASYNC_B128(l16, g16, gsrc, 16384);
    ASYNC_B128(l16, g16, gsrc, 20480);
    ASYNC_B128(l16, g16, gsrc, 24576);
    ASYNC_B128(l16, g16, gsrc, 28672);
    ASYNC_B128(l16, g16, gsrc, 32768);
    ASYNC_B128(l16, g16, gsrc, 36864);
    ASYNC_B128(l16, g16, gsrc, 40960);
    asm volatile("s_wait_asynccnt 0x0" ::: "memory");
    __syncthreads();

    // ---------------- Per-thread compute: points tid and tid+256 -------------
    float* myA = &smem[tid * CH];
    float* myB = myA + THREADS * CH;        // +22528 B -> folds into DS imm offset
    const char* myc = (const char*)myA;

    const float conA = myA[0], xA0 = myA[1], xA1 = myA[2], xA2 = myA[3];
    const float conB = myB[0], xB0 = myB[1], xB1 = myB[2], xB2 = myB[3];
    myA[0] = xA0 + xA1 + xA2;               // xsum -> x-index-3 slot
    myB[0] = xB0 + xB1 + xB2;

    const float k1     = P->k1;
    const float k2     = P->k2;
    const float nthinv = P->nthinv;
    float accA = conA * P->coef0;
    float accB = conB * P->coef0;

    // 12 proliferation Hill terms: p0 * (1 - p1/theta) * p2/(K1+p2)
    #pragma unroll
    for (int t = 0; t < N_PROLIF; ++t) {
        const int o0 = P->p_xoff0[t], o1 = P->p_xoff1[t], o2 = P->p_poff[t];
        const float cf = P->p_coef[t];
        {   const float p0 = *(const float*)(myc + o0);
            const float p1 = *(const float*)(myc + o1);
            const float p2 = *(const float*)(myc + o2);
            const float r  = __builtin_amdgcn_rcpf(k1 + p2);
            const float q  = fmaf(p1, nthinv, 1.0f);
            accA = fmaf(cf, (p0 * p2) * q * r, accA); }
        {   const float p0 = *(const float*)(myc + o0 + HALF_B);
            const float p1 = *(const float*)(myc + o1 + HALF_B);
            const float p2 = *(const float*)(myc + o2 + HALF_B);
            const float r  = __builtin_amdgcn_rcpf(k1 + p2);
            const float q  = fmaf(p1, nthinv, 1.0f);
            accB = fmaf(cf, (p0 * p2) * q * r, accB); }
    }

    // 18 linear Hill terms: l0 * |dp|/(K2+|dp|) * sigmoid(500*sign*dp - 25)
    #pragma unroll
    for (int t = 0; t < N_LIN; ++t) {
        const int o0 = P->l_xoff[t], op = P->l_poff[t];
        const float ssv = P->l_ss[t], sc = P->l_sc[t], cf = P->l_coef[t];
        {   const float l0  = *(const float*)(myc + o0);
            const float dp  = *(const float*)(myc + op) - ssv;
            const float adp = fabsf(dp);
            const float r   = __builtin_amdgcn_rcpf(k2 + adp);
#if USE_TANH
            const float s1  = __builtin_amdgcn_tanhf(fmaf(dp, sc, -12.5f)) + 1.0f;
            accA = fmaf(cf, (l0 * adp) * r * s1, accA);     // cf pre-halved
#else
            const float z   = fmaf(dp, sc, -25.0f);
            const float den = (k2 + adp) * (1.0f + __expf(-z));
            accA = fmaf(cf, (l0 * adp) * __builtin_amdgcn_rcpf(den), accA);
            (void)r;
#endif
        }
        {   const float l0  = *(const float*)(myc + o0 + HALF_B);
            const float dp  = *(const float*)(myc + op + HALF_B) - ssv;
            const float adp = fabsf(dp);
            const float r   = __builtin_amdgcn_rcpf(k2 + adp);
#if USE_TANH
            const float s1  = __builtin_amdgcn_tanhf(fmaf(dp, sc, -12.5f)) + 1.0f;
            accB = fmaf(cf, (l0 * adp) * r * s1, accB);
#else
            const float z   = fmaf(dp, sc, -25.0f);
            const float den = (k2 + adp) * (1.0f + __expf(-z));
            accB = fmaf(cf, (l0 * adp) * __builtin_amdgcn_rcpf(den), accB);
            (void)r;
#endif
        }
    }

    const size_t base = (size_t)blockIdx.x * PPB + tid;
    out[base]           = accA;      // coalesced b32 stores
    out[base + THREADS] = accB;
}

extern "C" void kernel_launch(void* const* d_in, const int* in_sizes, int n_in,
                              void* d_out, int out_size, void* d_ws, size_t ws_size,
                              hipStream_t stream) {
    const float* cand  = (const float*)d_in[0];
    const float* a     = (const float*)d_in[1];
    const float* ss    = (const float*)d_in[2];
    const float* sgn   = (const float*)d_in[3];
    const float* K1    = (const float*)d_in[4];
    const float* theta = (const float*)d_in[5];
    const float* K2    = (const float*)d_in[6];
    const int*   pidx  = (const int*)d_in[7];
    const int*   lidx  = (const int*)d_in[8];
    const unsigned char* sp = (const unsigned char*)d_in[9]; // jnp.bool_ = 1 byte
    float* out = (float*)d_out;
    Params* P  = (Params*)d_ws;

    setup_params<<<1, 32, 0, stream>>>(a, ss, sgn, K1, theta, K2, pidx, lidx, sp, P);

    // out_size = B*T = 2,097,152 = 4096 * 512 (exact)
    const int blocks = out_size / PPB;
    sindy_main<<<blocks, THREADS, 0, stream>>>(cand, P, out);
}